// GCN_4698694222362
// MI455X (gfx1250) — compile-verified
//
#include <hip/hip_runtime.h>
#include <math.h>

typedef __attribute__((ext_vector_type(2))) float v2f;
typedef __attribute__((ext_vector_type(8))) float v8f;

#define D_IN      128
#define D_H       64
#define D_OUT     40
#define D_OUT_PAD 48
#define XPAD      132   // 16-row x tile row pitch (bank stride 4, 16B aligned)
#define HPAD      68    // 16-row h tile row pitch (bank stride 4, 16B aligned)

// ---------- degree / normalization ----------
__global__ void k_init_deg(float* deg, int n) {
    int i = blockIdx.x * blockDim.x + threadIdx.x;
    if (i < n) deg[i] = 1.0f;                       // self-loop
}

__global__ void k_deg_accum(const long long* __restrict__ ei, float* deg, int E) {
    int e = blockIdx.x * blockDim.x + threadIdx.x;
    if (e < E) atomicAdd(&deg[(int)ei[(size_t)E + e]], 1.0f);
}

__global__ void k_dis(float* deg, int n) {
    int i = blockIdx.x * blockDim.x + threadIdx.x;
    if (i < n) deg[i] = rsqrtf(deg[i]);             // deg >= 1 always (self-loops)
}

// ---------- GEMM1: xw1[N,64] = x[N,128] @ W1[128,64]  (f32 WMMA 16x16x4) ----------
__global__ __launch_bounds__(64) void k_gemm1(const float* __restrict__ x,
                                              const float* __restrict__ W,
                                              float* __restrict__ xw, int n) {
    __shared__ float sW[D_IN * D_H];        // 32 KB, shared by both waves
    __shared__ float sX[2][16 * XPAD];      // 16x128 tile per wave, padded rows
    const int lane = threadIdx.x & 31;
    const int wave = threadIdx.x >> 5;

    // cooperative W1 load (float4)
    for (int i = threadIdx.x; i < (D_IN * D_H) / 4; i += 64)
        ((float4*)sW)[i] = ((const float4*)W)[i];

    const int rowTiles = (n + 15) >> 4;
    const int tile = blockIdx.x * 2 + wave;
    const bool active = tile < rowTiles * 4;
    const int rt = tile >> 2, ct = tile & 3;
    const int r0 = rt * 16, c0 = ct * 16;

    if (active) {
        const float4* x4 = (const float4*)x;
        for (int i = lane; i < 16 * (D_IN / 4); i += 32) {     // 512 float4s
            int row = i >> 5, cv = i & 31;                     // 32 float4 per row
            int grow = r0 + row; if (grow > n - 1) grow = n - 1;
            *(float4*)&sX[wave][row * XPAD + cv * 4] = x4[(size_t)grow * (D_IN / 4) + cv];
        }
    }
    __syncthreads();
    if (!active) return;                                       // whole-wave uniform

    const int koff = (lane < 16) ? 0 : 2;
    const int mn = lane & 15;
    v8f c = {};
    for (int k0 = 0; k0 < D_IN; k0 += 4) {
        v2f a, b;
        a.x = sX[wave][mn * XPAD + k0 + koff];
        a.y = sX[wave][mn * XPAD + k0 + koff + 1];
        b.x = sW[(k0 + koff) * D_H + c0 + mn];
        b.y = sW[(k0 + koff + 1) * D_H + c0 + mn];
        c = __builtin_amdgcn_wmma_f32_16x16x4_f32(false, a, false, b, (short)0, c, false, false);
    }
    const int rbase = (lane < 16) ? 0 : 8;
    for (int r = 0; r < 8; ++r) {
        int grow = r0 + rbase + r;
        if (grow < n) xw[(size_t)grow * D_H + c0 + mn] = c[r];
    }
}

// ---------- GEMM2: hw2[N,40] = h[N,64] @ W2[64,40]  (cols padded to 48) ----------
__global__ __launch_bounds__(64) void k_gemm2(const float* __restrict__ h,
                                              const float* __restrict__ W,
                                              float* __restrict__ hw, int n) {
    __shared__ float sW[D_H * D_OUT_PAD];   // 64x48, zero-padded cols 40..47
    __shared__ float sH[2][16 * HPAD];
    const int lane = threadIdx.x & 31;
    const int wave = threadIdx.x >> 5;

    for (int i = threadIdx.x; i < D_H * D_OUT_PAD; i += 64) {
        int k = i / D_OUT_PAD, cc = i % D_OUT_PAD;
        sW[i] = (cc < D_OUT) ? W[k * D_OUT + cc] : 0.0f;
    }

    const int rowTiles = (n + 15) >> 4;
    const int tile = blockIdx.x * 2 + wave;
    const bool active = tile < rowTiles * 3;
    const int rt = tile / 3, ct = tile % 3;
    const int r0 = rt * 16, c0 = ct * 16;

    if (active) {
        const float4* h4 = (const float4*)h;
        for (int i = lane; i < 16 * (D_H / 4); i += 32) {      // 256 float4s
            int row = i >> 4, cv = i & 15;                     // 16 float4 per row
            int grow = r0 + row; if (grow > n - 1) grow = n - 1;
            *(float4*)&sH[wave][row * HPAD + cv * 4] = h4[(size_t)grow * (D_H / 4) + cv];
        }
    }
    __syncthreads();
    if (!active) return;

    const int koff = (lane < 16) ? 0 : 2;
    const int mn = lane & 15;
    v8f c = {};
    for (int k0 = 0; k0 < D_H; k0 += 4) {
        v2f a, b;
        a.x = sH[wave][mn * HPAD + k0 + koff];
        a.y = sH[wave][mn * HPAD + k0 + koff + 1];
        b.x = sW[(k0 + koff) * D_OUT_PAD + c0 + mn];
        b.y = sW[(k0 + koff + 1) * D_OUT_PAD + c0 + mn];
        c = __builtin_amdgcn_wmma_f32_16x16x4_f32(false, a, false, b, (short)0, c, false, false);
    }
    const int rbase = (lane < 16) ? 0 : 8;
    const int col = c0 + mn;
    for (int r = 0; r < 8; ++r) {
        int grow = r0 + rbase + r;
        if (grow < n && col < D_OUT) hw[(size_t)grow * D_OUT + col] = c[r];
    }
}

// ---------- self-loop message init + edge scatter (layer 1) ----------
__global__ void k_self1(const float* __restrict__ xw, const float* __restrict__ dis,
                        float* __restrict__ acc, int n) {
    int t = blockIdx.x * blockDim.x + threadIdx.x;
    if (t < n * D_H) { float d = dis[t / D_H]; acc[t] = d * d * xw[t]; }
}

__global__ void k_scatter1(const long long* __restrict__ ei, const float* __restrict__ dis,
                           const float* __restrict__ xw, float* __restrict__ acc, int E) {
    int t = blockIdx.x * blockDim.x + threadIdx.x;
    if (t >= E * 16) return;
    int e = t >> 4, q = t & 15;                      // 16 float4 chunks per 64-wide row
    int s = (int)ei[e], d = (int)ei[(size_t)E + e];
    float w = dis[s] * dis[d];
    float4 v = ((const float4*)xw)[(size_t)s * 16 + q];
    float* dst = acc + (size_t)d * D_H + q * 4;
    atomicAdd(dst + 0, w * v.x);
    atomicAdd(dst + 1, w * v.y);
    atomicAdd(dst + 2, w * v.z);
    atomicAdd(dst + 3, w * v.w);
}

__global__ void k_relu_bias(float* __restrict__ acc, const float* __restrict__ b, int n) {
    int t = blockIdx.x * blockDim.x + threadIdx.x;
    if (t < n * D_H) acc[t] = fmaxf(acc[t] + b[t % D_H], 0.0f);
}

// ---------- layer 2 scatter into d_out ----------
__global__ void k_self2(const float* __restrict__ hw, const float* __restrict__ dis,
                        float* __restrict__ out, int n) {
    int t = blockIdx.x * blockDim.x + threadIdx.x;
    if (t < n * D_OUT) { float d = dis[t / D_OUT]; out[t] = d * d * hw[t]; }
}

__global__ void k_scatter2(const long long* __restrict__ ei, const float* __restrict__ dis,
                           const float* __restrict__ hw, float* __restrict__ out, int E) {
    int t = blockIdx.x * blockDim.x + threadIdx.x;
    if (t >= E * 10) return;
    int e = t / 10, q = t % 10;                      // 10 float4 chunks per 40-wide row
    int s = (int)ei[e], d = (int)ei[(size_t)E + e];
    float w = dis[s] * dis[d];
    float4 v = ((const float4*)hw)[(size_t)s * 10 + q];
    float* dst = out + (size_t)d * D_OUT + q * 4;
    atomicAdd(dst + 0, w * v.x);
    atomicAdd(dst + 1, w * v.y);
    atomicAdd(dst + 2, w * v.z);
    atomicAdd(dst + 3, w * v.w);
}

// ---------- bias + log_softmax, one wave32 per row, in place ----------
__global__ void k_lsm(float* __restrict__ out, const float* __restrict__ b, int n) {
    int row = blockIdx.x * 8 + (threadIdx.x >> 5);
    if (row >= n) return;
    int lane = threadIdx.x & 31;
    float* r = out + (size_t)row * D_OUT;
    float v0 = r[lane] + b[lane];                                   // lanes 0..31 < 40
    bool has1 = (lane + 32) < D_OUT;                                // lanes 0..7
    float v1 = has1 ? (r[lane + 32] + b[lane + 32]) : -3.402823e38f;
    float m = fmaxf(v0, v1);
    for (int off = 16; off > 0; off >>= 1) m = fmaxf(m, __shfl_xor(m, off, 32));
    float s = expf(v0 - m) + (has1 ? expf(v1 - m) : 0.0f);
    for (int off = 16; off > 0; off >>= 1) s += __shfl_xor(s, off, 32);
    float lse = logf(s);
    r[lane] = v0 - m - lse;
    if (has1) r[lane + 32] = v1 - m - lse;
}

extern "C" void kernel_launch(void* const* d_in, const int* in_sizes, int n_in,
                              void* d_out, int out_size, void* d_ws, size_t ws_size,
                              hipStream_t stream) {
    (void)n_in; (void)out_size; (void)ws_size;
    const float*     x  = (const float*)d_in[0];
    const long long* ei = (const long long*)d_in[1];   // int64 edge_index [2,E]
    const float*     W1 = (const float*)d_in[2];
    const float*     b1 = (const float*)d_in[3];
    const float*     W2 = (const float*)d_in[4];
    const float*     b2 = (const float*)d_in[5];
    float* out = (float*)d_out;

    const int n = in_sizes[0] / D_IN;
    const int E = in_sizes[1] / 2;

    float* dis  = (float*)d_ws;                 // N      (deg, then deg^-1/2 in place)
    float* xw1  = dis  + n;                     // N*64
    float* acc1 = xw1  + (size_t)n * D_H;       // N*64   (becomes h after bias+relu)
    float* hw2  = acc1 + (size_t)n * D_H;       // N*40

    const int B = 256;
    k_init_deg <<<(n + B - 1) / B, B, 0, stream>>>(dis, n);
    k_deg_accum<<<(E + B - 1) / B, B, 0, stream>>>(ei, dis, E);
    k_dis      <<<(n + B - 1) / B, B, 0, stream>>>(dis, n);

    const int rowTiles = (n + 15) / 16;
    const int tiles1 = rowTiles * 4;            // 64 / 16 column tiles
    k_gemm1<<<(tiles1 + 1) / 2, 64, 0, stream>>>(x, W1, xw1, n);

    k_self1   <<<((size_t)n * D_H + B - 1) / B, B, 0, stream>>>(xw1, dis, acc1, n);
    k_scatter1<<<((size_t)E * 16 + B - 1) / B, B, 0, stream>>>(ei, dis, xw1, acc1, E);
    k_relu_bias<<<((size_t)n * D_H + B - 1) / B, B, 0, stream>>>(acc1, b1, n);

    const int tiles2 = rowTiles * 3;            // 48 / 16 column tiles (40 padded)
    k_gemm2<<<(tiles2 + 1) / 2, 64, 0, stream>>>(acc1, W2, hw2, n);

    k_self2   <<<((size_t)n * D_OUT + B - 1) / B, B, 0, stream>>>(hw2, dis, out, n);
    k_scatter2<<<((size_t)E * 10 + B - 1) / B, B, 0, stream>>>(ei, dis, hw2, out, E);

    k_lsm<<<(n + 7) / 8, 256, 0, stream>>>(out, b2, n);
}